// MoE_876173328887
// MI455X (gfx1250) — compile-verified
//
#include <hip/hip_runtime.h>
#include <hip/hip_bf16.h>

typedef __attribute__((ext_vector_type(2))) float v2f;
typedef __attribute__((ext_vector_type(8))) float v8f;

#define KCOMP 4
#define HW 320
#define PIXELS (HW * HW)            // 102400
#define AROWS 96                    // 32 * 3
#define AKCOLS (AROWS * KCOMP)      // 384
#define AKTILES (AKCOLS / 16)       // 24
#define PTILES (PIXELS / 16)        // 6400
#define PT_PER_WAVE 8
#define PGROUPS (PTILES / PT_PER_WAVE)   // 800
#define TOTAL_WAVES (AKTILES * PGROUPS)  // 19200
#define BLOCK 256
#define WAVES_PER_BLOCK (BLOCK / 32)     // 8
#define NBLOCKS (TOTAL_WAVES / WAVES_PER_BLOCK) // 2400

// D = A(16 ak x 6 feat) * B(6 feat x 16 pixels), two K=4 WMMA steps, fp32 throughout.
// Per-lane after WMMA: 8 consecutive ak rows (2 full Gaussian groups) for one pixel,
// so Σe / Σwe / normalize / clamp are all in-register; stores are 16-lane coalesced.
__global__ __launch_bounds__(BLOCK) void moe_wmma_kernel(const float* __restrict__ params,
                                                         float* __restrict__ out) {
    const int lane = threadIdx.x & 31;
    const int wid  = blockIdx.x * WAVES_PER_BLOCK + (threadIdx.x >> 5);
    const int ak_tile = wid / PGROUPS;      // 0..23
    const int pgroup  = wid % PGROUPS;      // 0..799
    const int akbase  = ak_tile * 16;
    const int half    = lane >> 4;          // 0: lanes 0-15, 1: lanes 16-31
    const int l15     = lane & 15;

    // ---- A fragments: lane l15 owns ak = akbase + l15 (M = l15) ----
    const int ak   = akbase + l15;
    const int arow = ak >> 2;               // 0..95
    const int kc   = ak & 3;
    const float* pr = params + arow * 28;
    const float mux = pr[kc];
    const float muy = pr[4 + kc];
    const float s00 = pr[12 + 4 * kc + 0];
    const float s10 = pr[12 + 4 * kc + 2];
    const float s11 = pr[12 + 4 * kc + 3];
    // q = sA*dx^2 + sC*dx*dy + sB*dy^2 ; expand in (x,y), fold the -0.5 of exp(-q/2)
    const float sA = s00 * s00;
    const float sB = s10 * s10 + s11 * s11;
    const float sC = 2.0f * s00 * s10;
    const float c0 = -0.5f * sA;                                    // * x^2
    const float c1 = -0.5f * sB;                                    // * y^2
    const float c2 = -0.5f * sC;                                    // * x*y
    const float c3 =  0.5f * (2.0f * sA * mux + sC * muy);          // * x
    const float c4 =  0.5f * (2.0f * sB * muy + sC * mux);          // * y
    const float c5 = -0.5f * (sA * mux * mux + sB * muy * muy + sC * mux * muy); // * 1

    // A layout (32-bit 16x4): VGPR0 = {K0 | K2}, VGPR1 = {K1 | K3} per half-wave.
    v2f afrag0, afrag1;
    afrag0.x = half ? c2 : c0;
    afrag0.y = half ? c3 : c1;
    afrag1.x = half ? 0.0f : c4;
    afrag1.y = half ? 0.0f : c5;

    // ---- weights for this lane's 8 D rows (ak_local = half*8 + i) ----
    const int wr0 = akbase + half * 8;
    float wv[8];
#pragma unroll
    for (int i = 0; i < 8; ++i) {
        const int akw = wr0 + i;
        wv[i] = params[(akw >> 2) * 28 + 8 + (akw & 3)];
    }
    const int a0 = wr0 >> 2;   // output row for D rows 0..3 of this lane
    const int a1 = a0 + 1;     // output row for D rows 4..7

    const float inv = 1.0f / (float)HW;

    for (int t = 0; t < PT_PER_WAVE; ++t) {
        const int ptile = pgroup * PT_PER_WAVE + t;
        const int pix   = ptile * 16 + l15;          // this lane's pixel (N = l15)
        const int py    = pix / HW;
        const int px    = pix - py * HW;
        const float x = ((float)px + 0.5f) * inv;
        const float y = ((float)py + 0.5f) * inv;

        // B layout mirrors A: VGPR0 = {K0 | K2}, VGPR1 = {K1 | K3}.
        v2f bfrag0, bfrag1;
        bfrag0.x = half ? x * y : x * x;   // K2=xy | K0=x^2
        bfrag0.y = half ? x     : y * y;   // K3=x  | K1=y^2
        bfrag1.x = half ? 0.0f  : y;       //       | K0=y
        bfrag1.y = half ? 0.0f  : 1.0f;    //       | K1=1

        v8f acc = {};
        acc = __builtin_amdgcn_wmma_f32_16x16x4_f32(false, afrag0, false, bfrag0,
                                                    (short)0, acc, false, false);
        acc = __builtin_amdgcn_wmma_f32_16x16x4_f32(false, afrag1, false, bfrag1,
                                                    (short)0, acc, false, false);

        float e[8];
#pragma unroll
        for (int i = 0; i < 8; ++i) e[i] = __expf(acc[i]);

        const float g0 = e[0] + e[1] + e[2] + e[3];
        const float n0 = wv[0] * e[0] + wv[1] * e[1] + wv[2] * e[2] + wv[3] * e[3];
        const float g1 = e[4] + e[5] + e[6] + e[7];
        const float n1 = wv[4] * e[4] + wv[5] * e[5] + wv[6] * e[6] + wv[7] * e[7];

        // Fast reciprocal (v_rcp_f32, ~1 ulp) instead of the IEEE div sequence;
        // denominator is clamped to >= 1e-7 and the result is clamped to [0,1].
        float y0 = n0 * __builtin_amdgcn_rcpf(fmaxf(g0, 1e-7f));
        float y1 = n1 * __builtin_amdgcn_rcpf(fmaxf(g1, 1e-7f));
        y0 = fminf(fmaxf(y0, 0.0f), 1.0f);
        y1 = fminf(fmaxf(y1, 0.0f), 1.0f);

        out[a0 * PIXELS + pix] = y0;
        out[a1 * PIXELS + pix] = y1;
    }
}

extern "C" void kernel_launch(void* const* d_in, const int* in_sizes, int n_in,
                              void* d_out, int out_size, void* d_ws, size_t ws_size,
                              hipStream_t stream) {
    // inputs: d_in[0]=height(1), d_in[1]=width(1), d_in[2]=params(96*28 f32)
    (void)in_sizes; (void)n_in; (void)out_size; (void)d_ws; (void)ws_size;
    const float* params = (const float*)d_in[2];
    float* out = (float*)d_out;
    moe_wmma_kernel<<<NBLOCKS, BLOCK, 0, stream>>>(params, out);
}